// MultiHeadAttention_32049045963121
// MI455X (gfx1250) — compile-verified
//
#include <hip/hip_runtime.h>
#include <hip/hip_bf16.h>
#include <math.h>

typedef __attribute__((ext_vector_type(2))) float v2f;
typedef __attribute__((ext_vector_type(8))) float v8f;

static __device__ __forceinline__ v8f v8zero() {
  v8f z = {0.f, 0.f, 0.f, 0.f, 0.f, 0.f, 0.f, 0.f};
  return z;
}

// D = A(16x4) * B(4x16) + C, fp32 WMMA (keeps reference precision)
static __device__ __forceinline__ v8f wmma_f32(v2f a, v2f b, v8f c) {
  return __builtin_amdgcn_wmma_f32_16x16x4_f32(false, a, false, b, (short)0, c,
                                               false, false);
}

// ---------------------------------------------------------------------------
// Generic C[M,N] = A[M,K] @ B[N,K]^T + bias[N]   (torch Linear layout)
// workgroup tile 128(M) x 64(N), 8 waves of 32x32, K-chunks of 32 via LDS.
// ---------------------------------------------------------------------------
#define TK 32
#define LDS_AS 36   // 32 + 4 pad: 36*l mod 64 distinct for l=0..15, multiple of 4
__global__ __launch_bounds__(256) void gemm_abt_bias(
    const float* __restrict__ A, const float* __restrict__ B,
    const float* __restrict__ bias, float* __restrict__ C,
    int M, int N, int K) {
  __shared__ float As[128 * LDS_AS];
  __shared__ float Bs[64 * LDS_AS];
  const int tid = threadIdx.x;
  const int lane = tid & 31;
  const int wave = tid >> 5;
  const int l16 = lane & 15;
  const int hi = lane >> 4;            // 0: K cols 0..1, 1: K cols 2..3
  const int m0 = blockIdx.y * 128;
  const int n0 = blockIdx.x * 64;
  const int wm = (wave & 3) * 32;      // wave M offset
  const int wn = (wave >> 2) * 32;     // wave N offset

  v8f acc[2][2];
  for (int i = 0; i < 2; ++i)
    for (int j = 0; j < 2; ++j) acc[i][j] = v8zero();

  for (int k0 = 0; k0 < K; k0 += TK) {
    // cooperative coalesced loads into LDS
    {
      const int r = tid >> 3;          // 0..31
      const int cg = (tid & 7) * 4;    // 0..28
#pragma unroll
      for (int it = 0; it < 4; ++it) {
        const int row = r + it * 32;
        float4 v = *(const float4*)(A + (size_t)(m0 + row) * K + k0 + cg);
        *(float4*)(&As[row * LDS_AS + cg]) = v;
      }
#pragma unroll
      for (int it = 0; it < 2; ++it) {
        const int row = r + it * 32;
        float4 v = *(const float4*)(B + (size_t)(n0 + row) * K + k0 + cg);
        *(float4*)(&Bs[row * LDS_AS + cg]) = v;
      }
    }
    // prefetch next K-chunk (global_prefetch_b8)
    if (k0 + TK < K) {
      if (tid < 128)
        __builtin_prefetch(&A[(size_t)(m0 + tid) * K + k0 + TK], 0, 0);
      else if (tid < 192)
        __builtin_prefetch(&B[(size_t)(n0 + (tid - 128)) * K + k0 + TK], 0, 0);
    }
    __syncthreads();
#pragma unroll
    for (int kk = 0; kk < TK; kk += 4) {
      const int kc = kk + hi * 2;
      v2f a0 = *(const v2f*)(&As[(wm + l16) * LDS_AS + kc]);
      v2f a1 = *(const v2f*)(&As[(wm + 16 + l16) * LDS_AS + kc]);
      v2f b0 = *(const v2f*)(&Bs[(wn + l16) * LDS_AS + kc]);
      v2f b1 = *(const v2f*)(&Bs[(wn + 16 + l16) * LDS_AS + kc]);
      acc[0][0] = wmma_f32(a0, b0, acc[0][0]);
      acc[0][1] = wmma_f32(a0, b1, acc[0][1]);
      acc[1][0] = wmma_f32(a1, b0, acc[1][0]);
      acc[1][1] = wmma_f32(a1, b1, acc[1][1]);
    }
    __syncthreads();
  }
  // store with bias (C layout: VGPR j -> M=j (lanes<16) / j+8, lane -> N)
#pragma unroll
  for (int i = 0; i < 2; ++i) {
#pragma unroll
    for (int j = 0; j < 2; ++j) {
      const int col = n0 + wn + j * 16 + l16;
      const float bv = bias[col];
#pragma unroll
      for (int r = 0; r < 8; ++r) {
        const int row = m0 + wm + i * 16 + hi * 8 + r;
        C[(size_t)row * N + col] = acc[i][j][r] + bv;
      }
    }
  }
}

// ---------------------------------------------------------------------------
// Column logsumexp over the query axis:
// L[h,t] = log( t + sum_{s>=t} exp(qk[s,t]/8) )
// grid (128 t-blocks of 16, H heads); 8 waves split s-blocks round-robin.
// ---------------------------------------------------------------------------
__global__ __launch_bounds__(256) void col_logsumexp(
    const float* __restrict__ Qp, const float* __restrict__ Kp,
    float* __restrict__ L) {
  const int h = blockIdx.y;
  const int t0 = blockIdx.x * 16;
  const int tid = threadIdx.x;
  const int lane = tid & 31;
  const int wave = tid >> 5;
  const int l16 = lane & 15;
  const int hi = lane >> 4;
  __shared__ float redM[16][16];
  __shared__ float redS[16][16];

  // preload B frags: rows t0..t0+15 of Kp for this head (K=64 -> 16 k-steps)
  const float* kbase = Kp + (size_t)(t0 + l16) * 1024 + h * 64 + hi * 2;
  v2f bfrag[16];
#pragma unroll
  for (int kk = 0; kk < 16; ++kk) bfrag[kk] = *(const v2f*)(kbase + kk * 4);

  float mrun = -1e30f, srun = 0.f;
  const int imin = t0 >> 4;
  for (int i = imin + wave; i < 128; i += 8) {
    const int s0 = i * 16;
    const float* qbase = Qp + (size_t)(s0 + l16) * 1024 + h * 64 + hi * 2;
    v8f accv = v8zero();
#pragma unroll
    for (int kk = 0; kk < 16; ++kk) {
      v2f a = *(const v2f*)(qbase + kk * 4);
      accv = wmma_f32(a, bfrag[kk], accv);
    }
    const int t = t0 + l16;
    const bool full = (s0 >= t0 + 15);  // tile entirely below diagonal
    float xs[8];
    float tmax = -1e30f;
#pragma unroll
    for (int r = 0; r < 8; ++r) {
      const int s = s0 + hi * 8 + r;
      float x = accv[r] * 0.125f;
      if (!full && t > s) x = -INFINITY;  // excluded (handled by +t term)
      xs[r] = x;
      tmax = fmaxf(tmax, x);
    }
    const float mnew = fmaxf(mrun, tmax);
    float add = 0.f;
#pragma unroll
    for (int r = 0; r < 8; ++r) add += __expf(xs[r] - mnew);
    srun = srun * __expf(mrun - mnew) + add;
    mrun = mnew;
  }
  redM[l16][wave * 2 + hi] = mrun;
  redS[l16][wave * 2 + hi] = srun;
  __syncthreads();
  if (tid < 16) {
    float m = -1e30f, s = 0.f;
#pragma unroll
    for (int p = 0; p < 16; ++p) {
      const float m2 = redM[tid][p], s2 = redS[tid][p];
      const float mn = fmaxf(m, m2);
      s = s * __expf(m - mn) + s2 * __expf(m2 - mn);
      m = mn;
    }
    const int t = t0 + tid;
    if (t > 0) {  // t masked rows each contribute exp(0)=1
      const float mn = fmaxf(m, 0.f);
      s = s * __expf(m - mn) + (float)t * __expf(0.f - mn);
      m = mn;
    }
    L[h * 2048 + t] = m + __logf(s);
  }
}

// ---------------------------------------------------------------------------
// c[h,j] = sum_t L[h,t] * Vp[t, h*64+j]   (rank-1 correction, tiny)
// ---------------------------------------------------------------------------
__global__ __launch_bounds__(256) void lv_combine(
    const float* __restrict__ L, const float* __restrict__ Vp,
    float* __restrict__ cvec) {
  const int h = blockIdx.x;
  const int j = threadIdx.x & 63;
  const int part = threadIdx.x >> 6;
  __shared__ float red[4][64];
  float acc = 0.f;
  for (int t = part; t < 2048; t += 4)
    acc += L[h * 2048 + t] * Vp[(size_t)t * 1024 + h * 64 + j];
  red[part][j] = acc;
  __syncthreads();
  if (part == 0)
    cvec[h * 64 + j] = red[0][j] + red[1][j] + red[2][j] + red[3][j];
}

// ---------------------------------------------------------------------------
// Fused causal  Z[:, h*64:] = tril(q k^T)/8 @ v - c[h,:]
// grid (16 s-blocks of 128, H heads); wave owns 16 s-rows x 64 v-cols.
// Score tiles never leave the WGP (registers -> per-wave LDS transpose).
// ---------------------------------------------------------------------------
#define KV_LDS 68  // 64+4: 4n mod 64 distinct for n<16; halves land in disjoint bank classes
#define P_LDS 20
__global__ __launch_bounds__(256) void attn_av(
    const float* __restrict__ Qp, const float* __restrict__ Kp,
    const float* __restrict__ Vp, const float* __restrict__ cvec,
    float* __restrict__ Z) {
  const int h = blockIdx.y;
  const int s0 = blockIdx.x * 128;
  const int tid = threadIdx.x;
  const int lane = tid & 31;
  const int wave = tid >> 5;
  const int l16 = lane & 15;
  const int hi = lane >> 4;
  const int sw = s0 + wave * 16;  // this wave's query rows

  __shared__ float Ks[16 * KV_LDS];
  __shared__ float Vs[16 * KV_LDS];
  __shared__ float Ps[8][16 * P_LDS];

  // preload q frags for the wave's 16 rows (K=64 -> 16 A-frags)
  const float* qbase = Qp + (size_t)(sw + l16) * 1024 + h * 64 + hi * 2;
  v2f qfrag[16];
#pragma unroll
  for (int kk = 0; kk < 16; ++kk) qfrag[kk] = *(const v2f*)(qbase + kk * 4);

  v8f acc[4];
  for (int vt = 0; vt < 4; ++vt) acc[vt] = v8zero();

  const int ntb = (s0 + 128) >> 4;  // causal: t-blocks with t0 <= max s
  for (int tb = 0; tb < ntb; ++tb) {
    const int t0 = tb * 16;
    {  // cooperative load of K,V tiles (16 x 64 each), fully coalesced
      const int r = tid >> 4;
      const int cg = (tid & 15) * 4;
      *(float4*)(&Ks[r * KV_LDS + cg]) =
          *(const float4*)(Kp + (size_t)(t0 + r) * 1024 + h * 64 + cg);
      *(float4*)(&Vs[r * KV_LDS + cg]) =
          *(const float4*)(Vp + (size_t)(t0 + r) * 1024 + h * 64 + cg);
    }
    __syncthreads();
    if (sw + 15 >= t0) {  // wave-uniform: this wave has unmasked work here
      // P = q_blk @ k_blk^T  (16x16)
      v8f p = v8zero();
#pragma unroll
      for (int kk = 0; kk < 16; ++kk) {
        v2f b = *(const v2f*)(&Ks[l16 * KV_LDS + kk * 4 + hi * 2]);
        p = wmma_f32(qfrag[kk], b, p);
      }
      // mask (tril -> zero) + scale, stash in per-wave LDS for layout change
      const bool diag = (sw < t0 + 15);
#pragma unroll
      for (int r = 0; r < 8; ++r) {
        const int s = sw + hi * 8 + r;
        const int t = t0 + l16;
        float x = p[r] * 0.125f;
        if (diag && t > s) x = 0.f;
        Ps[wave][(hi * 8 + r) * P_LDS + l16] = x;
      }
      // acc += P @ V   (A-frags re-read from LDS; same-wave DS ops are in-order)
#pragma unroll
      for (int q4 = 0; q4 < 4; ++q4) {
        v2f af = *(const v2f*)(&Ps[wave][l16 * P_LDS + q4 * 4 + hi * 2]);
#pragma unroll
        for (int vt = 0; vt < 4; ++vt) {
          const int nc = vt * 16 + l16;
          v2f bf;
          bf.x = Vs[(q4 * 4 + hi * 2 + 0) * KV_LDS + nc];
          bf.y = Vs[(q4 * 4 + hi * 2 + 1) * KV_LDS + nc];
          acc[vt] = wmma_f32(af, bf, acc[vt]);
        }
      }
    }
    __syncthreads();
  }
  // Z[s, h*64+col] = acc - c[h,col]
#pragma unroll
  for (int vt = 0; vt < 4; ++vt) {
    const int col = vt * 16 + l16;
    const float cv = cvec[h * 64 + col];
#pragma unroll
    for (int r = 0; r < 8; ++r) {
      const int row = sw + hi * 8 + r;
      Z[(size_t)row * 1024 + h * 64 + col] = acc[vt][r] - cv;
    }
  }
}

// ---------------------------------------------------------------------------
extern "C" void kernel_launch(void* const* d_in, const int* in_sizes, int n_in,
                              void* d_out, int out_size, void* d_ws,
                              size_t ws_size, hipStream_t stream) {
  (void)in_sizes; (void)n_in; (void)out_size; (void)ws_size;
  const float* Qin = (const float*)d_in[0];
  const float* Kin = (const float*)d_in[1];
  const float* Vin = (const float*)d_in[2];
  const float* WQw = (const float*)d_in[3];
  const float* WQb = (const float*)d_in[4];
  const float* WKw = (const float*)d_in[5];
  const float* WKb = (const float*)d_in[6];
  const float* WVw = (const float*)d_in[7];
  const float* WVb = (const float*)d_in[8];
  const float* WOw = (const float*)d_in[9];
  const float* WOb = (const float*)d_in[10];
  float* out = (float*)d_out;

  float* ws = (float*)d_ws;                 // needs ~33.7 MB
  float* Qp = ws;                           // [2048,1024]
  float* Kp = Qp + 2048 * 1024;             // [2048,1024]
  float* Vp = Kp + 2048 * 1024;             // [2048,1024]
  float* Z  = Vp + 2048 * 1024;             // [2048,1024]
  float* L  = Z + 2048 * 1024;              // [16,2048]
  float* cv = L + 16 * 2048;                // [16,64]

  dim3 blk(256);
  dim3 gproj(1024 / 64, 2048 / 128);
  // q/k/v projections: stacked [H,64,1024] weights are a flat [1024,1024] W
  gemm_abt_bias<<<gproj, blk, 0, stream>>>(Qin, WQw, WQb, Qp, 2048, 1024, 1024);
  gemm_abt_bias<<<gproj, blk, 0, stream>>>(Kin, WKw, WKb, Kp, 2048, 1024, 1024);
  gemm_abt_bias<<<gproj, blk, 0, stream>>>(Vin, WVw, WVb, Vp, 2048, 1024, 1024);
  // column (query-axis) logsumexp per head
  col_logsumexp<<<dim3(128, 16), blk, 0, stream>>>(Qp, Kp, L);
  // rank-1 correction c = L @ V per head
  lv_combine<<<dim3(16), blk, 0, stream>>>(L, Vp, cv);
  // fused causal tril(QK^T)/8 @ V - c  ->  Z in [S, H*64] head-major layout
  attn_av<<<dim3(16, 16), blk, 0, stream>>>(Qp, Kp, Vp, cv, Z);
  // output projection
  gemm_abt_bias<<<gproj, blk, 0, stream>>>(Z, WOw, WOb, out, 2048, 1024, 1024);
}